// ShuffledGroupWhitening_6090263626199
// MI455X (gfx1250) — compile-verified
//
#include <hip/hip_runtime.h>

// ---------------- problem constants ----------------
#define NROWS 16384
#define DIM   2048
#define NG    64
#define GD    32          // per-group dim
#define EPSJ  1e-4f
#define NS_IT 12

typedef float v2f __attribute__((ext_vector_type(2)));
typedef float v8f __attribute__((ext_vector_type(8)));

// V_WMMA_F32_16X16X4_F32: D(16x16 f32) = A(16x4 f32) * B(4x16 f32) + C
// 8-arg pattern per probe: (neg_a, A, neg_b, B, c_mod, C, reuse_a, reuse_b)
static __device__ __forceinline__ v8f wmma_f32_4(v2f a, v2f b, v8f c) {
  return __builtin_amdgcn_wmma_f32_16x16x4_f32(false, a, false, b, (short)0, c,
                                               false, false);
}

// =====================================================================
// Kernel 1: per-(group, row-slice) raw second moment + column sums.
// grid = (64, 8), block = 256 (8 waves). Each wave: 256 rows, 64 steps of K=4.
// A-tile layout (16x4 f32): lanes 0-15 = M rows, VGPR0/1 = K0/K1;
// lanes 16-31 = same M, K2/K3.  B (4x16) mirrors with N in lanes.
// For C = Xg^T Xg both operands are the same register data.
// =====================================================================
__global__ __launch_bounds__(256) void sgw_stats(
    const float* __restrict__ x, const long long* __restrict__ perm,
    float* __restrict__ part_cov /*[512][1024]*/,
    float* __restrict__ part_sum /*[512][32]*/) {
  const int g    = blockIdx.x;
  const int by   = blockIdx.y;
  const int tid  = threadIdx.x;
  const int wave = tid >> 5;
  const int lane = tid & 31;
  const int half = lane >> 4;   // 0: K=0,1 rows; 1: K=2,3 rows
  const int lid  = lane & 15;

  __shared__ int   s_cols[GD];
  __shared__ float s_red[8 * 1024];          // per-wave cov partials (32 KB)
  __shared__ float s_sum[8][2][2][16];       // per-wave column-sum partials

  if (tid < GD) s_cols[tid] = (int)perm[g * GD + tid];
  __syncthreads();

  const int c0 = s_cols[lid];        // group col lid
  const int c1 = s_cols[lid + 16];   // group col lid+16

  const int rbase = (by * 8 + wave) * 256;
  const float* p0 = x + (size_t)(rbase + half * 2) * DIM;

  v8f a00 = {0.f,0.f,0.f,0.f,0.f,0.f,0.f,0.f};
  v8f a01 = a00, a10 = a00, a11 = a00;
  float s0 = 0.f, s1 = 0.f;

#pragma unroll 2
  for (int it = 0; it < 64; ++it) {
    const float* pr = p0 + (size_t)it * 4 * DIM;   // rows rbase+4*it+2*half{,+1}
    float v00 = pr[c0];
    float v01 = pr[DIM + c0];
    float v10 = pr[c1];
    float v11 = pr[DIM + c1];
    s0 += v00 + v01;
    s1 += v10 + v11;
    v2f t0 = {v00, v01};   // cols 0..15 slice  (valid as both A and B operand)
    v2f t1 = {v10, v11};   // cols 16..31 slice
    a00 = wmma_f32_4(t0, t0, a00);
    a01 = wmma_f32_4(t0, t1, a01);
    a10 = wmma_f32_4(t1, t0, a10);
    a11 = wmma_f32_4(t1, t1, a11);
  }

  // scatter wave-partials to LDS.  C layout: VGPR i -> M=i(+8 for hi lanes), N=lane
  {
    float* dst = &s_red[wave * 1024];
#pragma unroll
    for (int i = 0; i < 8; ++i) {
      int d0 = i + half * 8;
      dst[(d0) * 32      + lid]      = a00[i];
      dst[(d0) * 32      + lid + 16] = a01[i];
      dst[(d0 + 16) * 32 + lid]      = a10[i];
      dst[(d0 + 16) * 32 + lid + 16] = a11[i];
    }
    s_sum[wave][half][0][lid] = s0;
    s_sum[wave][half][1][lid] = s1;
  }
  __syncthreads();

  // deterministic cross-wave reduce -> global partials
  for (int idx = tid; idx < 1024; idx += 256) {
    float v = 0.f;
#pragma unroll
    for (int w = 0; w < 8; ++w) v += s_red[w * 1024 + idx];
    part_cov[(size_t)(g * 8 + by) * 1024 + idx] = v;
  }
  if (tid < 32) {
    float v = 0.f;
#pragma unroll
    for (int w = 0; w < 8; ++w)
      for (int h = 0; h < 2; ++h) v += s_sum[w][h][tid >> 4][tid & 15];
    part_sum[(size_t)(g * 8 + by) * 32 + tid] = v;
  }
}

// =====================================================================
// Kernel 2: reduce partials, C = M2/N - mu mu^T + eps I
// =====================================================================
__global__ __launch_bounds__(256) void sgw_finalize(
    const float* __restrict__ part_cov, const float* __restrict__ part_sum,
    float* __restrict__ mean, float* __restrict__ cov) {
  const int g = blockIdx.x, tid = threadIdx.x;
  __shared__ float s_mean[GD];
  if (tid < GD) {
    float s = 0.f;
#pragma unroll
    for (int b = 0; b < 8; ++b) s += part_sum[(size_t)(g * 8 + b) * 32 + tid];
    float m = s * (1.0f / NROWS);
    s_mean[tid] = m;
    mean[g * GD + tid] = m;
  }
  __syncthreads();
  for (int idx = tid; idx < 1024; idx += 256) {
    float v = 0.f;
#pragma unroll
    for (int b = 0; b < 8; ++b) v += part_cov[(size_t)(g * 8 + b) * 1024 + idx];
    int d = idx >> 5, e = idx & 31;
    float c = v * (1.0f / NROWS) - s_mean[d] * s_mean[e] + ((d == e) ? EPSJ : 0.f);
    cov[(size_t)g * 1024 + idx] = c;
  }
}

// =====================================================================
// Kernel 3: coupled Newton-Schulz inverse sqrt of each 32x32 covariance.
// A_n = A/tr(A) (PSD -> eigs in (0,1]); Y0=A_n, Z0=I;
// T=(3I-ZY)/2; Y<-YT; Z<-TZ; Z -> A_n^{-1/2};  W = Z / sqrt(tr).
// =====================================================================
__global__ __launch_bounds__(256) void sgw_ns_invsqrt(
    const float* __restrict__ cov, float* __restrict__ W) {
  const int g = blockIdx.x, tid = threadIdx.x;
  __shared__ float A[1024], Y[1024], Z[1024], T[1024], S[1024];
  __shared__ float tr;

  for (int idx = tid; idx < 1024; idx += 256) A[idx] = cov[(size_t)g * 1024 + idx];
  __syncthreads();
  if (tid == 0) {
    float t = 0.f;
    for (int d = 0; d < GD; ++d) t += A[d * 32 + d];
    tr = t;
  }
  __syncthreads();
  const float itr = 1.0f / tr;
  for (int idx = tid; idx < 1024; idx += 256) {
    Y[idx] = A[idx] * itr;
    Z[idx] = ((idx >> 5) == (idx & 31)) ? 1.0f : 0.0f;
  }
  __syncthreads();

  for (int it = 0; it < NS_IT; ++it) {
    for (int idx = tid; idx < 1024; idx += 256) {
      int r = idx >> 5, c = idx & 31;
      float acc = 0.f;
#pragma unroll
      for (int k = 0; k < 32; ++k) acc += Z[r * 32 + k] * Y[k * 32 + c];
      T[idx] = ((r == c) ? 1.5f : 0.f) - 0.5f * acc;
    }
    __syncthreads();
    for (int idx = tid; idx < 1024; idx += 256) {
      int r = idx >> 5, c = idx & 31;
      float ay = 0.f, az = 0.f;
#pragma unroll
      for (int k = 0; k < 32; ++k) {
        ay += Y[r * 32 + k] * T[k * 32 + c];
        az += T[r * 32 + k] * Z[k * 32 + c];
      }
      S[idx] = ay;   // new Y
      A[idx] = az;   // new Z (A is free after init)
    }
    __syncthreads();
    for (int idx = tid; idx < 1024; idx += 256) { Y[idx] = S[idx]; Z[idx] = A[idx]; }
    __syncthreads();
  }

  const float isq = rsqrtf(tr);
  for (int idx = tid; idx < 1024; idx += 256)
    W[(size_t)g * 1024 + idx] = Z[idx] * isq;
}

// =====================================================================
// Kernel 4: y = (xg - mu) * W, scattered back through perm.
// grid = (64 groups, 128 row-blocks of 128), block = 256 (8 waves).
// Each wave: 16 rows x 32 output cols = 2 C-tiles, K=32 in 8 WMMA steps.
// =====================================================================
__global__ __launch_bounds__(256) void sgw_apply(
    const float* __restrict__ x, const long long* __restrict__ perm,
    const float* __restrict__ mean, const float* __restrict__ Wm,
    float* __restrict__ out) {
  const int g    = blockIdx.x;
  const int rb   = blockIdx.y;
  const int tid  = threadIdx.x;
  const int wave = tid >> 5;
  const int lane = tid & 31;
  const int half = lane >> 4;
  const int lid  = lane & 15;

  __shared__ int   s_cols[GD];
  __shared__ float s_mean[GD];
  __shared__ float s_W[1024];

  if (tid < GD) {
    s_cols[tid] = (int)perm[g * GD + tid];
    s_mean[tid] = mean[g * GD + tid];
  }
  for (int idx = tid; idx < 1024; idx += 256) s_W[idx] = Wm[(size_t)g * 1024 + idx];
  __syncthreads();

  const int rowt = rb * 128 + wave * 16;       // tile row base
  const int row  = rowt + lid;                 // this lane's A row (M=lid)
  const float* xr = x + (size_t)row * DIM;

  // lane's 16 centered values: K = 4*kk + 2*half + {0,1}
  float a[16];
#pragma unroll
  for (int kk = 0; kk < 8; ++kk) {
    int k0 = 4 * kk + 2 * half;
    a[2 * kk]     = xr[s_cols[k0]]     - s_mean[k0];
    a[2 * kk + 1] = xr[s_cols[k0 + 1]] - s_mean[k0 + 1];
  }

  v8f c0 = {0.f,0.f,0.f,0.f,0.f,0.f,0.f,0.f};
  v8f c1 = c0;
#pragma unroll
  for (int kk = 0; kk < 8; ++kk) {
    int k0 = 4 * kk + 2 * half;
    v2f av = {a[2 * kk], a[2 * kk + 1]};
    v2f b0 = {s_W[k0 * 32 + lid],       s_W[(k0 + 1) * 32 + lid]};
    v2f b1 = {s_W[k0 * 32 + 16 + lid],  s_W[(k0 + 1) * 32 + 16 + lid]};
    c0 = wmma_f32_4(av, b0, c0);
    c1 = wmma_f32_4(av, b1, c1);
  }

  // store through perm (un-shuffle): C VGPR i -> row rowt+i(+8 hi lanes), f=lid(+16)
  const int oc0 = s_cols[lid];
  const int oc1 = s_cols[lid + 16];
#pragma unroll
  for (int i = 0; i < 8; ++i) {
    int ro = rowt + i + half * 8;
    out[(size_t)ro * DIM + oc0] = c0[i];
    out[(size_t)ro * DIM + oc1] = c1[i];
  }
}

// =====================================================================
extern "C" void kernel_launch(void* const* d_in, const int* in_sizes, int n_in,
                              void* d_out, int out_size, void* d_ws, size_t ws_size,
                              hipStream_t stream) {
  const float*     x    = (const float*)d_in[0];
  const long long* perm = (const long long*)d_in[1];   // int64 per reference
  float* out = (float*)d_out;

  float* ws       = (float*)d_ws;
  float* part_cov = ws;                      // 512*1024
  float* part_sum = part_cov + 512 * 1024;   // 512*32
  float* mean     = part_sum + 512 * 32;     // 64*32
  float* cov      = mean + NG * GD;          // 64*1024
  float* Wm       = cov + NG * 1024;         // 64*1024
  (void)in_sizes; (void)n_in; (void)out_size; (void)ws_size;

  sgw_stats   <<<dim3(NG, 8),   256, 0, stream>>>(x, perm, part_cov, part_sum);
  sgw_finalize<<<dim3(NG),      256, 0, stream>>>(part_cov, part_sum, mean, cov);
  sgw_ns_invsqrt<<<dim3(NG),    256, 0, stream>>>(cov, Wm);
  sgw_apply   <<<dim3(NG, 128), 256, 0, stream>>>(x, perm, mean, Wm, out);
}